// Adder2D_71433896067355
// MI455X (gfx1250) — compile-verified
//
#include <hip/hip_runtime.h>
#include <hip/hip_bf16.h>
#include <stdint.h>

// Problem constants (fixed by the reference harness)
constexpr int TPB   = 256;     // 8 waves of 32
constexpr int N_    = 8;
constexpr int CI    = 64;
constexpr int H     = 32;
constexpr int W     = 32;
constexpr int CO    = 64;
constexpr int K     = 3;
constexpr int HO    = 32;
constexpr int WO    = 32;

constexpr int CI_CHUNK = 8;                 // ci planes staged per DMA chunk
constexpr int NCHUNK   = CI / CI_CHUNK;     // 8
constexpr int RSTR     = 40;                // padded LDS row stride (floats) -> 16B-aligned interior
constexpr int PROWS    = 34;                // 32 interior rows + 2 halo rows
constexpr int PLANE    = PROWS * RSTR;      // 1360 floats / plane
constexpr int XBUF     = CI_CHUNK * PLANE;  // 10880 floats / buffer
constexpr int CO_PER_WG= 4;
constexpr int WSLICE   = CO_PER_WG * CI * K * K; // 2304 floats

// LDS byte offset of a __shared__ object: flat LDS address truncates to the
// in-aperture offset on gfx1250 (addr[31:0] == lds offset).
__device__ __forceinline__ uint32_t lds_off(const void* p) {
  return (uint32_t)(uintptr_t)p;
}

// CDNA5 async global->LDS DMA (ASYNCcnt-tracked), CDNA5 ISA
__device__ __forceinline__ void async_ld_b128(uint32_t loff, const void* g) {
  asm volatile("global_load_async_to_lds_b128 %0, %1, off"
               :: "v"(loff), "v"(g) : "memory");
}
__device__ __forceinline__ void async_ld_b32(uint32_t loff, const void* g) {
  asm volatile("global_load_async_to_lds_b32 %0, %1, off"
               :: "v"(loff), "v"(g) : "memory");
}

__global__ __launch_bounds__(TPB)
void adder2d_cdna5_kernel(const float* __restrict__ x,
                          const float* __restrict__ w,
                          float* __restrict__ out) {
  // 2 x-staging buffers (double buffered) + full w slice for this WG's 4 co's.
  __shared__ float xs[2][XBUF];   // 87,040 B
  __shared__ float wl[WSLICE];    //  9,216 B

  const int t   = threadIdx.x;
  const int n   = blockIdx.x >> 4;   // CO/CO_PER_WG == 16 groups
  const int cog = blockIdx.x & 15;

  // ---- zero LDS x buffers (establishes the zero halo for padding==1) ----
  {
    float* xsf = &xs[0][0];
    for (int i = t; i < 2 * XBUF; i += TPB) xsf[i] = 0.0f;
  }
  __syncthreads();   // drain DS stores in every wave before DMA touches LDS

  const float* xg = x + (size_t)n * (CI * H * W);
  const float* wg = w + (size_t)(cog * CO_PER_WG) * (CI * K * K);

  // ---- async-load entire w slice once: 2304 floats, 9 b32 per thread ----
#pragma unroll
  for (int i = 0; i < 9; ++i) {
    const int idx = t + i * TPB;                 // 0..2303, layout matches wl
    async_ld_b32(lds_off(&wl[idx]), wg + idx);
  }

  // ---- async-load x chunk 0 into buffer 0 (8 b128 per thread) ----
  {
    const int row  = t >> 3;          // 0..31
    const int col4 = (t & 7) << 2;    // 0,4,...,28
#pragma unroll
    for (int ci = 0; ci < CI_CHUNK; ++ci) {
      async_ld_b128(lds_off(&xs[0][ci * PLANE + (row + 1) * RSTR + 4 + col4]),
                    xg + ci * (H * W) + row * W + col4);
    }
  }

  // ---- per-thread output assignment: one co, one 4x4 pixel tile ----
  const int co_l = t >> 6;            // 0..3
  const int tile = t & 63;            // 64 tiles of 4x4 covering 32x32
  const int oy0  = (tile >> 3) * 4;
  const int ox0  = (tile & 7) * 4;

  float acc[16];
#pragma unroll
  for (int i = 0; i < 16; ++i) acc[i] = 0.0f;

  // ---- main pipeline over ci chunks (double-buffered DMA / compute) ----
  for (int c = 0; c < NCHUNK; ++c) {
    if (c + 1 < NCHUNK) {
      // prefetch next chunk into the other buffer before draining this one
      const int row  = t >> 3;
      const int col4 = (t & 7) << 2;
      float* xb = &xs[(c + 1) & 1][0];
      const float* gsrc = xg + (size_t)(c + 1) * CI_CHUNK * (H * W);
#pragma unroll
      for (int ci = 0; ci < CI_CHUNK; ++ci) {
        async_ld_b128(lds_off(&xb[ci * PLANE + (row + 1) * RSTR + 4 + col4]),
                      gsrc + ci * (H * W) + row * W + col4);
      }
      // 8 newer per-wave async ops in flight -> older chunk (+w) complete
      asm volatile("s_wait_asynccnt 8" ::: "memory");
    } else {
      asm volatile("s_wait_asynccnt 0" ::: "memory");
    }
    __syncthreads();  // make every wave's DMA-landed data visible WG-wide

    const float* xb = &xs[c & 1][0];
#pragma unroll
    for (int ci = 0; ci < CI_CHUNK; ++ci) {
      // 6x6 x patch feeding a 4x4 output tile through 3x3 taps
      const float* xp = xb + ci * PLANE + oy0 * RSTR + 3 + ox0;
      float xv[6][6];
#pragma unroll
      for (int r = 0; r < 6; ++r)
#pragma unroll
        for (int cc = 0; cc < 6; ++cc)
          xv[r][cc] = xp[r * RSTR + cc];

      const float* wp = &wl[co_l * (CI * K * K) + (c * CI_CHUNK + ci) * 9];
      float wv[9];
#pragma unroll
      for (int k = 0; k < 9; ++k) wv[k] = wp[k];   // LDS broadcast across lanes

#pragma unroll
      for (int py = 0; py < 4; ++py)
#pragma unroll
        for (int px = 0; px < 4; ++px) {
          float a = acc[py * 4 + px];
#pragma unroll
          for (int di = 0; di < 3; ++di)
#pragma unroll
            for (int dj = 0; dj < 3; ++dj)
              a += fabsf(xv[py + di][px + dj] - wv[di * 3 + dj]);
          acc[py * 4 + px] = a;
        }
    }
    __syncthreads();  // compute done before next iteration's DMA reuses buffer
  }

  // ---- store: out = -acc, float4 per tile row ----
  const int co = cog * CO_PER_WG + co_l;
  float* op = out + ((size_t)(n * CO + co) * (HO * WO)) + oy0 * WO + ox0;
#pragma unroll
  for (int py = 0; py < 4; ++py) {
    float4 v = make_float4(-acc[py * 4 + 0], -acc[py * 4 + 1],
                           -acc[py * 4 + 2], -acc[py * 4 + 3]);
    *reinterpret_cast<float4*>(op + py * WO) = v;
  }
}

extern "C" void kernel_launch(void* const* d_in, const int* in_sizes, int n_in,
                              void* d_out, int out_size, void* d_ws, size_t ws_size,
                              hipStream_t stream) {
  const float* x = (const float*)d_in[0];  // [8,64,32,32]
  const float* w = (const float*)d_in[1];  // [64,64,3,3]
  float* out = (float*)d_out;              // [8,64,32,32]
  dim3 grid(N_ * (CO / CO_PER_WG));        // 128 workgroups
  adder2d_cdna5_kernel<<<grid, TPB, 0, stream>>>(x, w, out);
}